// RodriguesLoss_81277961109615
// MI455X (gfx1250) — compile-verified
//
#include <hip/hip_runtime.h>
#include <stdint.h>

// RodriguesLoss on MI455X (gfx1250).
// Bandwidth-bound: 153 MB in / 4 B out; ~7us HBM floor @23.3TB/s. No GEMM
// structure -> WMMA intentionally NOT used (matrix pipes can't beat the
// bandwidth bound; the only per-sample matmul is 17x3 * 3x3 = ~150 FLOP).
// CDNA5 features used: Tensor Data Mover (tensor_load_to_lds, TENSORcnt)
// for bulk tile staging -- one DMA descriptor per tile instead of ~4700
// per-lane loads -- with async global->LDS b128 as compile-time fallback;
// wave32 shuffle reduction; deterministic two-pass final reduce.

#define NT 256          // threads per block (8 waves)
#define NSB 256         // samples per block (1 per thread)
#define NANG 22
#define NCO 51

#if defined(__has_builtin)
#if __has_builtin(__builtin_amdgcn_tensor_load_to_lds)
#define HAVE_TDM 1
#endif
#endif
#ifndef HAVE_TDM
#define HAVE_TDM 0
#endif

#if HAVE_TDM
typedef unsigned int tdm_v4u __attribute__((ext_vector_type(4)));
typedef int          tdm_v8i __attribute__((ext_vector_type(8)));
typedef int          tdm_v4i __attribute__((ext_vector_type(4)));

// 1D contiguous DMA: N 4-byte elements, global -> LDS, as a 1-row 2D tensor.
// D# layout per CDNA5 ISA (groups 0/1; rest zero).
// This toolchain exposes the 6-arg builtin:
//   (uint32x4 g0, int32x8 g1, int32x4, int32x4, int32x8, i32 cpol)
__device__ __forceinline__ void tdm_copy_1d(uint32_t lds_off, const void* gsrc,
                                            uint32_t nElem) {
    uint64_t ga = (uint64_t)gsrc;
    tdm_v4u g0;
    g0[0] = 1u;                                   // count=1 (valid user D#)
    g0[1] = lds_off;                              // lds_addr (bytes)
    g0[2] = (uint32_t)ga;                         // global_addr[31:0]
    g0[3] = (uint32_t)(ga >> 32) | (2u << 30);    // global_addr[56:32] | type=2
    tdm_v8i g1;
    g1[0] = (int)(2u << 16);                      // data_size=2 -> 4 bytes
    g1[1] = (int)((nElem & 0xFFFFu) << 16);       // tensor_dim0[15:0]
    g1[2] = (int)((nElem >> 16) | (1u << 16));    // tensor_dim0[31:16] | tensor_dim1=1
    g1[3] = (int)(nElem << 16);                   // tile_dim0 = nElem (<= 65535)
    g1[4] = 1;                                    // tile_dim1 = 1, tile_dim2 = 0
    g1[5] = (int)nElem;                           // tensor_dim0_stride[31:0]
    g1[6] = 0;                                    // stride0 hi | stride1 lo
    g1[7] = 0;
    tdm_v4i z4 = {0, 0, 0, 0};                    // groups 2/3 unused (<=2D)
    tdm_v8i z8 = {0, 0, 0, 0, 0, 0, 0, 0};        // extra group (6-arg form)
    __builtin_amdgcn_tensor_load_to_lds(g0, g1, z4, z4, z8, 0);
}
#endif

struct F3 { float x, y, z; };
__device__ __forceinline__ F3 mk(float x, float y, float z) { return {x, y, z}; }
__device__ __forceinline__ F3 operator+(F3 a, F3 b) { return {a.x + b.x, a.y + b.y, a.z + b.z}; }
__device__ __forceinline__ F3 operator-(F3 a, F3 b) { return {a.x - b.x, a.y - b.y, a.z - b.z}; }
__device__ __forceinline__ F3 operator-(F3 a) { return {-a.x, -a.y, -a.z}; }
__device__ __forceinline__ F3 operator*(float s, F3 a) { return {s * a.x, s * a.y, s * a.z}; }
__device__ __forceinline__ float dot3(F3 a, F3 b) { return a.x * b.x + a.y * b.y + a.z * b.z; }
__device__ __forceinline__ F3 cross3(F3 a, F3 b) {
    return {a.y * b.z - a.z * b.y, a.z * b.x - a.x * b.z, a.x * b.y - a.y * b.x};
}
__device__ __forceinline__ F3 nrm(F3 v) {
    float inv = rsqrtf(dot3(v, v));
    return inv * v;
}
// Rodrigues: c*v + (1-c)*(v.k)*k + s*(k x v)
__device__ __forceinline__ F3 rod(F3 v, F3 k, float th) {
    float s, c;
    __sincosf(th, &s, &c);
    return c * v + ((1.0f - c) * dot3(v, k)) * k + s * cross3(k, v);
}
__device__ __forceinline__ float sqn(F3 d) { return d.x * d.x + d.y * d.y + d.z * d.z; }

__global__ __launch_bounds__(NT) void rodrigues_loss_kernel(
    const float* __restrict__ ang, const float* __restrict__ coord,
    float* __restrict__ partial, int nsamp)
{
    __shared__ float lsA[NSB * NANG];   // 22528 B
    __shared__ float lsC[NSB * NCO];    // 52224 B
    __shared__ float red[NT / 32];

    const int t = threadIdx.x;
    const long blockBase = (long)blockIdx.x * NSB;
    const int valid = min(NSB, nsamp - (int)blockBase);

    // ---- Bulk tile staging into LDS ----
    {
        uint32_t lA = (uint32_t)(uintptr_t)&lsA[0];  // LDS byte offset
        uint32_t lC = (uint32_t)(uintptr_t)&lsC[0];
#if HAVE_TDM
        // Tensor Data Mover: one descriptor per tile, issued by wave 0 only
        // (TDM ignores EXEC; per-wave issue; tracked by TENSORcnt).
        if (t < 32) {
            tdm_copy_1d(lA, ang + blockBase * NANG, (uint32_t)valid * NANG);
            tdm_copy_1d(lC, coord + blockBase * NCO, (uint32_t)valid * NCO);
            __builtin_amdgcn_s_wait_tensorcnt(0);
        }
#else
        // Fallback: coalesced per-lane async global->LDS b128 (ASYNCcnt).
        uint32_t aBytes = ((uint32_t)valid * NANG * 4u + 15u) & ~15u;
        uint32_t cBytes = ((uint32_t)valid * NCO * 4u + 15u) & ~15u;
        uint64_t gA = (uint64_t)(const void*)(ang + blockBase * NANG);
        uint64_t gC = (uint64_t)(const void*)(coord + blockBase * NCO);
        for (uint32_t off = (uint32_t)t * 16u; off < aBytes; off += NT * 16u) {
            uint32_t l = lA + off;
            uint64_t g = gA + off;
            asm volatile("global_load_async_to_lds_b128 %0, %1, off"
                         :: "v"(l), "v"(g) : "memory");
        }
        for (uint32_t off = (uint32_t)t * 16u; off < cBytes; off += NT * 16u) {
            uint32_t l = lC + off;
            uint64_t g = gC + off;
            asm volatile("global_load_async_to_lds_b128 %0, %1, off"
                         :: "v"(l), "v"(g) : "memory");
        }
        asm volatile("s_wait_asynccnt 0" ::: "memory");
#endif
    }
    __syncthreads();

    float loss = 0.0f;
    if (t < valid) {
        const float* a  = &lsA[t * NANG];
        const float* cw = &lsC[t * NCO];

        F3 cW[17];
#pragma unroll
        for (int i = 0; i < 17; ++i)
            cW[i] = mk(cw[3 * i], cw[3 * i + 1], cw[3 * i + 2]);

        // ---- Torso plane least-squares fit: solve (X^T X) xt = X^T z ----
        const int TOR[7] = {0, 1, 4, 7, 8, 11, 14};
        float Sxx = 0, Sxy = 0, Sx = 0, Syy = 0, Sy = 0, Sxz = 0, Syz = 0, Sz = 0;
#pragma unroll
        for (int i = 0; i < 7; ++i) {
            F3 p = cW[TOR[i]];
            Sxx += p.x * p.x; Sxy += p.x * p.y; Sx += p.x;
            Syy += p.y * p.y; Sy += p.y;
            Sxz += p.x * p.z; Syz += p.y * p.z; Sz += p.z;
        }
        const float S1 = 7.0f;
        float M00 = Syy * S1 - Sy * Sy;
        float M01 = Sxy * S1 - Sy * Sx;
        float M02 = Sxy * Sy - Syy * Sx;
        float det = Sxx * M00 - Sxy * M01 + Sx * M02;
        float inv = 1.0f / det;
        float t1  = Syz * S1 - Sz * Sy;
        float d0  = Sxz * M00 - Sxy * t1 + Sx * (Syz * Sy - Sz * Syy);
        float d1  = Sxx * t1 - Sxz * M01 + Sx * (Sxy * Sz - Syz * Sx);
        F3 xt = mk(d0 * inv, d1 * inv, -1.0f);   // xt[2] forced to -1 per reference

        F3 yt = cross3(cW[8] - cW[0], xt);
        F3 zt = cross3(xt, yt);
        xt = nrm(xt); yt = nrm(yt); zt = nrm(zt);

        F3 cT[17];
#pragma unroll
        for (int i = 0; i < 17; ++i)
            cT[i] = mk(dot3(xt, cW[i]), dot3(yt, cW[i]), dot3(zt, cW[i]));

        const F3 xT = mk(1, 0, 0), yT = mk(0, 1, 0), zT = mk(0, 0, 1);

        // Head
        F3 gt0 = nrm(cT[9] - cT[8]);
        F3 gt1 = nrm(cross3(gt0, cross3(cT[10] - cT[9], -gt0)));
        F3 vHP = rod(zT, yT, a[0]);
        F3 vHY = rod(rod(xT, yT, a[0]), vHP, a[1]);

        // Left arm
        F3 gt2 = nrm(cT[12] - cT[11]);
        F3 gt3 = nrm(cT[13] - cT[12]);
        F3 vLSP = rod(xT, yT, a[3]);
        F3 e811 = cT[8] - cT[11];
        F3 vLSR = rod(vLSP, nrm(cross3(e811, vLSP)), a[4]);
        F3 vLER = rod(vLSR, nrm(cross3(e811, vLSR)), a[5]);
        F3 vLEY = rod(vLER, vLSR, a[6]);

        // Right arm
        F3 gt4 = nrm(cT[15] - cT[14]);
        F3 gt5 = nrm(cT[16] - cT[15]);
        F3 vRSP = rod(xT, yT, a[6]);
        F3 e814 = cT[8] - cT[14];
        F3 vRSR = rod(vRSP, nrm(cross3(vRSP, e814)), a[7]);
        F3 vRER = rod(vRSR, nrm(cross3(vRSR, e814)), a[8]);
        F3 vREY = rod(vRER, vRSR, a[9]);

        const float is2 = 0.70710678118654752f;

        // Left leg
        F3 gt6 = nrm(cT[5] - cT[4]);
        F3 gt7 = nrm(cT[6] - cT[5]);
        F3 vLHP  = rod(-zT, yT, a[10]);
        F3 kLHR  = rod(xT, yT, a[10]);
        F3 kLHYP = rod(mk(0.0f, is2, -is2), yT, a[10]);
        F3 vLHR  = rod(vLHP, kLHR, a[11]);
        kLHYP    = rod(kLHYP, kLHR, a[11]);
        F3 vLHYP = rod(vLHR, kLHYP, a[12]);
        F3 d65 = cT[6] - cT[5];
        F3 vLKP = rod(vLHYP, nrm(cross3(d65, -vLHYP)), a[13]);

        // Right leg
        F3 gt8 = nrm(cT[2] - cT[1]);
        F3 gt9 = nrm(cT[3] - cT[2]);
        F3 vRHP  = rod(-zT, yT, a[14]);
        F3 kRHR  = rod(xT, yT, a[14]);
        F3 kRHYP = rod(mk(0.0f, is2, is2), yT, a[14]);
        F3 vRHR  = rod(vRHP, kRHR, a[15]);
        kRHYP    = rod(kRHYP, kRHR, a[15]);
        F3 vRHYP = rod(vRHR, kRHYP, a[16]);
        F3 d32 = cT[3] - cT[2];
        F3 vRKP = rod(vRHYP, nrm(cross3(d32, -vRHYP)), a[17]);

        // Ankles (keys rotated back to world: k @ R = k.x*xt + k.y*yt + k.z*zt)
        F3 kLAP = nrm(cross3(d65, cT[4] - cT[5]));
        F3 kLAR = nrm(cross3(kLAP, cT[5] - cT[6]));
        F3 kRAP = nrm(cross3(d32, cT[1] - cT[2]));
        F3 kRAR = nrm(cross3(kRAP, cT[2] - cT[3]));
        F3 kLAPW = kLAP.x * xt + kLAP.y * yt + kLAP.z * zt;
        F3 kLARW = kLAR.x * xt + kLAR.y * yt + kLAR.z * zt;
        F3 kRAPW = kRAP.x * xt + kRAP.y * yt + kRAP.z * zt;
        F3 kRARW = kRAR.x * xt + kRAR.y * yt + kRAR.z * zt;
        F3 vLAP = rod(kLARW, kLAPW, a[18]);
        F3 vLAR = rod(kLAPW, vLAP, a[19]);
        F3 vRAP = rod(kRARW, kRAPW, a[20]);
        F3 vRAR = rod(kRAPW, vRAP, a[21]);

        // Sum of squared diffs over the 14x3 stacks; gt rows 10..13 are zero
        // and sim rows 10..13 are masked by z_T (only z survives).
        float s = 0.0f;
        s += sqn(gt0 - vHP);
        s += sqn(gt1 - vHY);
        s += sqn(gt2 - vLSR);
        s += sqn(gt3 - vLEY);
        s += sqn(gt4 - vRSR);
        s += sqn(gt5 - vREY);
        s += sqn(gt6 - vLHYP);
        s += sqn(gt7 - vLKP);
        s += sqn(gt8 - vRHYP);
        s += sqn(gt9 - vRKP);
        s += vLAP.z * vLAP.z + vLAR.z * vLAR.z + vRAP.z * vRAP.z + vRAR.z * vRAR.z;
        loss = s;
    }

    // ---- wave32 shuffle reduction + cross-wave LDS combine ----
#pragma unroll
    for (int o = 16; o > 0; o >>= 1)
        loss += __shfl_down(loss, o, 32);
    if ((t & 31) == 0) red[t >> 5] = loss;
    __syncthreads();
    if (t == 0) {
        float b = 0.0f;
#pragma unroll
        for (int i = 0; i < NT / 32; ++i) b += red[i];
        partial[blockIdx.x] = b;
    }
}

__global__ __launch_bounds__(256) void rodrigues_reduce_kernel(
    const float* __restrict__ partial, int n, float* __restrict__ out, float scale)
{
    __shared__ float red[8];
    float s = 0.0f;
    for (int i = threadIdx.x; i < n; i += 256) s += partial[i];
#pragma unroll
    for (int o = 16; o > 0; o >>= 1)
        s += __shfl_down(s, o, 32);
    if ((threadIdx.x & 31) == 0) red[threadIdx.x >> 5] = s;
    __syncthreads();
    if (threadIdx.x == 0) {
        float b = 0.0f;
#pragma unroll
        for (int i = 0; i < 8; ++i) b += red[i];
        out[0] = b * scale;
    }
}

extern "C" void kernel_launch(void* const* d_in, const int* in_sizes, int n_in,
                              void* d_out, int out_size, void* d_ws, size_t ws_size,
                              hipStream_t stream) {
    const float* ang   = (const float*)d_in[0];   // (B, 22) f32
    const float* coord = (const float*)d_in[1];   // (B, 51) f32
    int n = in_sizes[0] / NANG;                    // B = 524288
    int nblocks = (n + NSB - 1) / NSB;             // 2048 partials -> d_ws
    float* partial = (float*)d_ws;

    rodrigues_loss_kernel<<<nblocks, NT, 0, stream>>>(ang, coord, partial, n);
    // final scale: mean over 14x3 elements (1/42) then mean over B (1/n)
    rodrigues_reduce_kernel<<<1, 256, 0, stream>>>(partial, nblocks, (float*)d_out,
                                                   1.0f / (42.0f * (float)n));
}